// DynamicDirectionalConv_3289944948773
// MI455X (gfx1250) — compile-verified
//
#include <hip/hip_runtime.h>

// MI455X / gfx1250, wave32. fp32 WMMA (V_WMMA_F32_16X16X4_F32) for both 1x1 convs,
// TDM (tensor_load_to_lds) for the interior x_low tile staging in kernel 2.
typedef float v2f __attribute__((ext_vector_type(2)));
typedef float v8f __attribute__((ext_vector_type(8)));
typedef unsigned int u32x4 __attribute__((ext_vector_type(4)));
typedef int i32x4 __attribute__((ext_vector_type(4)));
typedef int i32x8 __attribute__((ext_vector_type(8)));

#define BATCH 2
#define CH    256
#define HH    256
#define WW    256
#define HW    65536
#define MID   16
#define RW_P  260        // padded LDS pitch for reduce_w rows (260 % 64 == 4 -> conflict-free)
#define TS    16         // spatial tile for kernel 2
#define HALO  22         // TS + 2*3
#define XLP   (HALO*HALO)
#define OLP   18         // out_low LDS pitch (even -> 8B-aligned float2, stride-18 banks distinct)

// ---------------------------------------------------------------------------
// Kernel 1: x_low[b][m][p] = sum_c x[b][c][p] * reduce_w[m][c]
// One wave = one 16(mid) x 16(pixel) D tile, K=256 via 64 x wmma_f32_16x16x4_f32.
// ---------------------------------------------------------------------------
__global__ __launch_bounds__(256) void k_reduce(const float* __restrict__ x,
                                                const float* __restrict__ reduce_w,
                                                float* __restrict__ xlow) {
    __shared__ float lds_rw[MID * RW_P];
    const int tid = threadIdx.x;
    for (int i = tid; i < MID * CH; i += 256) {
        int m = i >> 8, c = i & 255;
        lds_rw[m * RW_P + c] = reduce_w[i];
    }
    __syncthreads();

    const int wid  = tid >> 5;
    const int lane = tid & 31;
    const int g    = blockIdx.x * 8 + wid;     // 16-pixel group in [0, B*H*W/16)
    const int p0   = g << 4;
    const int b    = p0 >> 16;                 // HW = 65536
    const int pimg = (p0 & (HW - 1)) + (lane & 15);

    const int m  = lane & 15;                  // A-frag M row
    const int ko = (lane >> 4) << 1;           // K sub-offset {0,2}

    const float* xb = x + (size_t)b * CH * HW + pimg;
    v8f acc = {};
    for (int ks = 0; ks < 64; ++ks) {
        const int k = (ks << 2) + ko;          // channel index for this lane's A/B pair
        v2f a = *(const v2f*)(&lds_rw[m * RW_P + k]);   // reduce_w[m][k], [m][k+1]
        v2f bf;
        bf.x = xb[(size_t)k * HW];             // x[b][k][pix]   (lanes 0-15: 64B contiguous)
        bf.y = xb[(size_t)(k + 1) * HW];
        acc = __builtin_amdgcn_wmma_f32_16x16x4_f32(false, a, false, bf,
                                                    (short)0, acc, false, false);
    }
    // D layout: VGPR r -> rows M=r (lanes 0-15) / M=r+8 (lanes 16-31), N = lane&15
    const int mrow = (lane >> 4) << 3;
    float* ob = xlow + (size_t)b * MID * HW + (p0 & (HW - 1)) + (lane & 15);
#pragma unroll
    for (int r = 0; r < 8; ++r)
        ob[(size_t)(mrow + r) * HW] = acc[r];
}

// ---------------------------------------------------------------------------
// TDM: 3D tile load 22(x) x 22(y) x 16(z) fp32, strides (1, W, H*W) -> LDS
// contiguous [z][y][x] == lds_x[m*484 + row*22 + col]. Descriptor per ISA 8.3/8.4.
// ---------------------------------------------------------------------------
__device__ __forceinline__ void tdm_load_tile(unsigned int lds_byte_off,
                                              const float* gsrc) {
    unsigned long long ga = (unsigned long long)(uintptr_t)gsrc;
    u32x4 g0;
    g0[0] = 1u;                                  // count=1 (valid user D#)
    g0[1] = lds_byte_off;                        // lds_addr
    g0[2] = (unsigned int)(ga & 0xffffffffu);    // global_addr[31:0]
    g0[3] = (unsigned int)((ga >> 32) & 0x01ffffffu) | (2u << 30); // addr[56:32], type=2
    i32x8 g1;
    g1[0] = 0x00020000;                          // data_size=2 (4B), no multicast/pad/iter
    g1[1] = (int)(22u << 16);                    // tensor_dim0[15:0]=22
    g1[2] = (int)(22u << 16);                    // tensor_dim0 hi=0, tensor_dim1[15:0]=22
    g1[3] = (int)(22u << 16);                    // tensor_dim1 hi=0, tile_dim0=22
    g1[4] = (int)(22u | (16u << 16));            // tile_dim1=22, tile_dim2=16
    g1[5] = (int)WW;                             // tensor_dim0_stride = 256
    g1[6] = 0;                                   // stride0 hi, stride1[15:0] (65536 lo = 0)
    g1[7] = 1;                                   // tensor_dim1_stride[47:16] = 65536>>16
    i32x4 g2;
    g2[0] = MID;                                 // tensor_dim2 = 16
    g2[1] = 0; g2[2] = 0; g2[3] = 0;             // dim3 / dim2_stride unused (tile_dim3=0)
    i32x4 g3 = {0, 0, 0, 0};
#if defined(__clang_major__) && (__clang_major__ >= 23)
    i32x8 g4 = {0, 0, 0, 0, 0, 0, 0, 0};
    __builtin_amdgcn_tensor_load_to_lds(g0, g1, g2, g3, g4, 0);
#else
    __builtin_amdgcn_tensor_load_to_lds(g0, g1, g2, g3, 0);
#endif
}

// ---------------------------------------------------------------------------
// Kernel 2: per 16x16 tile: TDM tile stage -> angle-MLP -> combined 7x7 kernel ->
// directional conv over 16 mids -> expand GEMM (WMMA) -> out[b][c][h][w].
// ---------------------------------------------------------------------------
__global__ __launch_bounds__(256) void k_dirconv_expand(
        const float* __restrict__ xlow, const float* __restrict__ angle,
        const float* __restrict__ w1, const float* __restrict__ b1,
        const float* __restrict__ w2, const float* __restrict__ b2,
        const float* __restrict__ expand_w, float* __restrict__ out) {
    __shared__ float lds_x[MID * XLP];          // halo'd x_low tile; reused for out_low
    __shared__ float lds_base[4 * 49];          // oriented Gaussian kernels
    __shared__ float lds_ew[CH * MID];          // expand_w [c][m]

    const int tid = threadIdx.x;
    const int bid = blockIdx.x;                 // 512 blocks: b*256 + th*16 + tw
    const int b  = bid >> 8;
    const int h0 = ((bid >> 4) & 15) << 4;
    const int w0 = (bid & 15) << 4;

    // Interior tiles: halo never crosses the image -> pure rectangular copy (TDM).
    // Border tiles: per-lane loads with reflect indexing. Branch is block-uniform,
    // so EXEC is all-1s at every WMMA below.
    const bool interior = (h0 >= 3) && (h0 + 19 <= HH) && (w0 >= 3) && (w0 + 19 <= WW);
    if (interior) {
        if (tid < 32) {                          // TDM ignores EXEC: issue from wave 0 only
            const float* gsrc = xlow + (size_t)(b * MID) * HW
                              + (size_t)(h0 - 3) * WW + (w0 - 3);
            tdm_load_tile((unsigned int)(uintptr_t)lds_x, gsrc);
        }
    } else {
        for (int i = tid; i < MID * XLP; i += 256) {
            const int m   = i / XLP;
            const int idx = i - m * XLP;
            const int hh  = idx / HALO, ww = idx - (idx / HALO) * HALO;
            int gh = h0 + hh - 3, gw = w0 + ww - 3;
            gh = gh < 0 ? -gh : (gh >= HH ? 2 * HH - 2 - gh : gh);
            gw = gw < 0 ? -gw : (gw >= WW ? 2 * WW - 2 - gw : gw);
            lds_x[i] = xlow[(size_t)(b * MID + m) * HW + gh * WW + gw];
        }
    }

    // 4 threads each build one normalized anisotropic Gaussian (sigma 2.5 / 1.0)
    if (tid < 4) {
        const float theta = 0.78539816339744831f * (float)tid;   // k * pi/4
        const float ct = __cosf(theta), st = __sinf(theta);
        float tmp[49], sum = 0.f;
#pragma unroll
        for (int i = 0; i < 7; ++i)
#pragma unroll
            for (int j = 0; j < 7; ++j) {
                const float xx = (float)(i - 3), yy = (float)(j - 3);
                const float xr = xx * ct + yy * st;
                const float yr = -xx * st + yy * ct;
                const float v = __expf(-(xr * xr * 0.08f + yr * yr * 0.5f));
                tmp[i * 7 + j] = v; sum += v;
            }
        const float inv = 1.f / sum;
#pragma unroll
        for (int t = 0; t < 49; ++t) lds_base[tid * 49 + t] = tmp[t] * inv;
    }
    for (int i = tid; i < CH * MID; i += 256) lds_ew[i] = expand_w[i];

    // --- per-pixel MLP -> softmax mixing weights (global reads only: overlaps TDM) ---
    const int li = tid >> 4, lj = tid & 15;
    const float a2 = 2.f * angle[(size_t)b * HW + (h0 + li) * WW + (w0 + lj)];
    const float f0 = __sinf(a2), f1 = __cosf(a2);
    float hb[8];
#pragma unroll
    for (int o = 0; o < 8; ++o) {
        const float v = w1[2 * o] * f0 + w1[2 * o + 1] * f1 + b1[o];
        hb[o] = v > 0.f ? v : 0.f;
    }
    float z[4];
#pragma unroll
    for (int k = 0; k < 4; ++k) {
        float v = b2[k];
#pragma unroll
        for (int o = 0; o < 8; ++o) v += w2[k * 8 + o] * hb[o];
        z[k] = v;
    }
    const float mz = fmaxf(fmaxf(z[0], z[1]), fmaxf(z[2], z[3]));
    const float e0 = __expf(z[0] - mz), e1 = __expf(z[1] - mz);
    const float e2 = __expf(z[2] - mz), e3 = __expf(z[3] - mz);
    const float si = 1.f / (e0 + e1 + e2 + e3);
    const float k0w = e0 * si, k1w = e1 * si, k2w = e2 * si, k3w = e3 * si;

    if (interior && tid < 32)                   // wave 0 owns the TENSORcnt
        __builtin_amdgcn_s_wait_tensorcnt(0);
    __syncthreads();

    // combined 49-tap kernel for this pixel (4x fewer conv FLOPs than 4 separate convs)
    float ker[49];
#pragma unroll
    for (int t = 0; t < 49; ++t)
        ker[t] = k0w * lds_base[t] + k1w * lds_base[49 + t]
               + k2w * lds_base[98 + t] + k3w * lds_base[147 + t];

    // --- directional conv: one pixel per thread, all 16 mids ---
    float ol[MID];
    for (int m = 0; m < MID; ++m) {
        float acc = 0.f;
        const float* xm = &lds_x[m * XLP + li * HALO + lj];
#pragma unroll
        for (int ti = 0; ti < 7; ++ti)
#pragma unroll
            for (int tj = 0; tj < 7; ++tj)
                acc += ker[ti * 7 + tj] * xm[ti * HALO + tj];
        ol[m] = acc;
    }
    __syncthreads();                           // everyone done reading lds_x

    float* lds_ol = lds_x;                     // reuse: out_low [pixel][OLP]
#pragma unroll
    for (int m2 = 0; m2 < MID; m2 += 2) {
        v2f t; t.x = ol[m2]; t.y = ol[m2 + 1];
        *(v2f*)(&lds_ol[tid * OLP + m2]) = t;
    }
    __syncthreads();

    // --- expand GEMM: D[c-tile 16][pixel 16] = expand_w(16xK) x out_low^T(Kx16) ---
    const int wid = tid >> 5, lane = tid & 31;
    const int laneM = lane & 15;
    const int ko = (lane >> 4) << 1;
    for (int t = 0; t < 32; ++t) {             // 8 waves x 32 = 16 pg x 16 c-tiles
        const int T  = wid * 32 + t;
        const int pg = T & 15;
        const int c0 = (T >> 4) << 4;
        v8f acc = {};
#pragma unroll
        for (int ks = 0; ks < 4; ++ks) {
            const int k = (ks << 2) + ko;
            v2f a  = *(const v2f*)(&lds_ew[(c0 + laneM) * MID + k]);  // expand_w[c][m]
            v2f bf = *(const v2f*)(&lds_ol[(pg * 16 + laneM) * OLP + k]);
            acc = __builtin_amdgcn_wmma_f32_16x16x4_f32(false, a, false, bf,
                                                        (short)0, acc, false, false);
        }
        // D: row c0+r (+8 for hi lanes), cols = 16 consecutive w -> 64B runs
        const size_t ob = ((size_t)b * CH + c0 + ((lane >> 4) << 3)) * HW
                        + (size_t)(h0 + pg) * WW + (w0 + laneM);
#pragma unroll
        for (int r = 0; r < 8; ++r)
            out[ob + (size_t)r * HW] = acc[r];
    }
}

extern "C" void kernel_launch(void* const* d_in, const int* in_sizes, int n_in,
                              void* d_out, int out_size, void* d_ws, size_t ws_size,
                              hipStream_t stream) {
    const float* x        = (const float*)d_in[0];
    const float* angle    = (const float*)d_in[1];
    const float* reduce_w = (const float*)d_in[2];
    const float* expand_w = (const float*)d_in[3];
    const float* w1       = (const float*)d_in[4];
    const float* b1       = (const float*)d_in[5];
    const float* w2       = (const float*)d_in[6];
    const float* b2       = (const float*)d_in[7];
    float* xlow = (float*)d_ws;                // B*MID*H*W*4 = 8.4 MB scratch
    float* out  = (float*)d_out;

    k_reduce<<<1024, 256, 0, stream>>>(x, reduce_w, xlow);          // 8192 pixel-groups
    k_dirconv_expand<<<512, 256, 0, stream>>>(xlow, angle, w1, b1, w2, b2,
                                              expand_w, out);       // 512 16x16 tiles
    (void)in_sizes; (void)n_in; (void)out_size; (void)ws_size;
}